// Graph_Layer_44787918963025
// MI455X (gfx1250) — compile-verified
//
#include <hip/hip_runtime.h>

typedef __attribute__((ext_vector_type(2))) float v2f;
typedef __attribute__((ext_vector_type(8))) float v8f;

#define NROW 8192
#define KDIM 512
#define EPS_ 1e-6f
#define DEPS2 (512.0f * EPS_ * EPS_)

__device__ __forceinline__ v8f wmma_f32(v2f a, v2f b, v8f c) {
  // D = A(16x4) * B(4x16) + C(16x16), all fp32
  return __builtin_amdgcn_wmma_f32_16x16x4_f32(false, a, false, b, (short)0, c,
                                               false, false);
}

// CDNA5 async global->LDS copy (16B per lane), tracked by ASYNCcnt.
// Generic LDS pointer low 32 bits == LDS byte offset (flat aperture rule),
// which is what the instruction's VDST operand wants.
__device__ __forceinline__ void async_copy_b128(float* lds_dst, const float* gsrc) {
  const unsigned lds_addr = (unsigned)(unsigned long long)lds_dst;
  const unsigned long long ga = (unsigned long long)gsrc;
  asm volatile("global_load_async_to_lds_b128 %0, %1, off"
               :
               : "v"(lds_addr), "v"(ga)
               : "memory");
}

__device__ __forceinline__ void wait_async() {
  asm volatile("s_wait_asynccnt 0x0" ::: "memory");
}

// Fragment from a row-major [row][k] LDS tile. Per ISA layout, both the A
// operand (M x K) and the B operand (K x N, source stored N-major with K
// contiguous, i.e. B = S^T) load identically:
//   lane L: row = base + (L&15), k = k0 + 2*(L>>4) + {0,1}
__device__ __forceinline__ v2f frag_rm(const float* base, int ld, int row_base, int k0) {
  const int lane = threadIdx.x & 31;
  const float* p = base + (row_base + (lane & 15)) * ld + k0 + ((lane >> 4) << 1);
  v2f r; r.x = p[0]; r.y = p[1];
  return r;
}

// B fragment from a k-major [k][n] LDS tile (x and weight are stored this way).
__device__ __forceinline__ v2f frag_km(const float* base, int ld, int k0, int n_base) {
  const int lane = threadIdx.x & 31;
  const float* p = base + (k0 + ((lane >> 4) << 1)) * ld + n_base + (lane & 15);
  v2f r; r.x = p[0]; r.y = p[ld];
  return r;
}

__device__ __forceinline__ float dist_val(float asq, float as, float bsq, float bs, float g) {
  const float t = asq + bsq - 2.0f * g + 2.0f * EPS_ * (as - bs) + DEPS2;
  return sqrtf(fmaxf(t, 1e-12f));
}

// ---------------------------------------------------------------------------
// Kernel 1: per-row sum of squares / sum of sim_feat; zero rowmax.
// ---------------------------------------------------------------------------
__global__ __launch_bounds__(256) void stats_init_kernel(
    const float* __restrict__ S, float* __restrict__ sq,
    float* __restrict__ ssum, float* __restrict__ rowmax) {
  const int row = blockIdx.x * 8 + (threadIdx.x >> 5);
  const int lane = threadIdx.x & 31;
  const float4* p = (const float4*)(S + (size_t)row * KDIM);
  float a2 = 0.0f, a1 = 0.0f;
#pragma unroll
  for (int i = 0; i < 4; ++i) {
    const float4 v = p[lane + 32 * i];
    a2 += v.x * v.x + v.y * v.y + v.z * v.z + v.w * v.w;
    a1 += v.x + v.y + v.z + v.w;
  }
#pragma unroll
  for (int m = 16; m >= 1; m >>= 1) {
    a2 += __shfl_xor(a2, m, 32);
    a1 += __shfl_xor(a1, m, 32);
  }
  if (lane == 0) {
    sq[row] = a2;
    ssum[row] = a1;
    rowmax[row] = 0.0f;  // bit pattern 0 == identity for uint atomicMax
  }
}

// ---------------------------------------------------------------------------
// Kernel 2: rowmax[i] = max_j D(i,j); gram via fp32 WMMA, never materialized.
// Block = 32 i-rows; loops j in chunks of 128; 8 waves, each owns a 16-row
// strip and two 16-col tiles. All staging via async global->LDS.
// ---------------------------------------------------------------------------
__global__ __launch_bounds__(256) void rowmax_kernel(
    const float* __restrict__ S, const float* __restrict__ sq,
    const float* __restrict__ ssum, float* __restrict__ rowmax) {
  extern __shared__ float lds[];
  float* smA = lds;          // 32 x 516 (S i-rows, K fully resident)
  float* smB = lds + 16512;  // 128 x 68 (S j-rows, 64-wide K chunk)
  float* sqi = lds + 25216;  // 32
  float* si  = lds + 25248;  // 32
  float* sqj = lds + 25280;  // 128
  float* sj  = lds + 25408;  // 128  (total 25536 floats = 102,144 B)

  const int t = threadIdx.x;
  const int wave = t >> 5, lane = t & 31;
  const int half = lane >> 4, ln = lane & 15;
  const int i0 = blockIdx.x * 32;

  for (int idx = t; idx < 32 * 128; idx += 256) {
    const int r = idx >> 7, c = idx & 127;
    async_copy_b128(smA + r * 516 + c * 4, S + (size_t)(i0 + r) * KDIM + c * 4);
  }
  if (t < 32) { sqi[t] = sq[i0 + t]; si[t] = ssum[i0 + t]; }
  wait_async();

  const int strip = wave & 1;
  const int m_base = strip * 16;
  const int c0 = wave >> 1;  // tiles c0 and c0+4 of the 8 col tiles

  float rmax[8] = {0.f, 0.f, 0.f, 0.f, 0.f, 0.f, 0.f, 0.f};

  for (int j0 = 0; j0 < NROW; j0 += 128) {
    __syncthreads();
    if (t < 128) { sqj[t] = sq[j0 + t]; sj[t] = ssum[j0 + t]; }
    v8f g0 = {}, g1 = {};
    for (int kc = 0; kc < KDIM; kc += 64) {
      __syncthreads();
      for (int idx = t; idx < 128 * 16; idx += 256) {
        const int r = idx >> 4, c = idx & 15;
        async_copy_b128(smB + r * 68 + c * 4,
                        S + (size_t)(j0 + r) * KDIM + kc + c * 4);
      }
      wait_async();
      __syncthreads();
#pragma unroll
      for (int k = 0; k < 64; k += 4) {
        const v2f a  = frag_rm(smA, 516, m_base, kc + k);
        const v2f b0 = frag_rm(smB, 68, c0 * 16, k);
        const v2f b1 = frag_rm(smB, 68, (c0 + 4) * 16, k);
        g0 = wmma_f32(a, b0, g0);
        g1 = wmma_f32(a, b1, g1);
      }
    }
    const float bsq0 = sqj[c0 * 16 + ln], bs0 = sj[c0 * 16 + ln];
    const float bsq1 = sqj[(c0 + 4) * 16 + ln], bs1 = sj[(c0 + 4) * 16 + ln];
#pragma unroll
    for (int v = 0; v < 8; ++v) {
      const int m = m_base + v + 8 * half;
      const float asq = sqi[m], as = si[m];
      const float d0 = dist_val(asq, as, bsq0, bs0, g0[v]);
      const float d1 = dist_val(asq, as, bsq1, bs1, g1[v]);
      rmax[v] = fmaxf(rmax[v], fmaxf(d0, d1));
    }
  }
#pragma unroll
  for (int v = 0; v < 8; ++v) {
#pragma unroll
    for (int m = 8; m >= 1; m >>= 1) rmax[v] = fmaxf(rmax[v], __shfl_xor(rmax[v], m, 32));
  }
  if (ln == 0) {
#pragma unroll
    for (int v = 0; v < 8; ++v)
      atomicMax((unsigned int*)&rowmax[i0 + m_base + v + 8 * half],
                __float_as_uint(rmax[v]));  // valid: all D >= 0
  }
}

// ---------------------------------------------------------------------------
// Kernel 3: Y = G @ x, with G tiles produced on the fly (gram via WMMA ->
// elementwise normalize -> G tile parked in LDS -> WMMA against x chunks).
// Block = 32 i-rows x all 512 x-cols (8 chunks of 64), j in chunks of 128.
// ---------------------------------------------------------------------------
__global__ __launch_bounds__(256) void fused_gx_kernel(
    const float* __restrict__ S, const float* __restrict__ X,
    const float* __restrict__ sq, const float* __restrict__ ssum,
    const float* __restrict__ rowmax, float* __restrict__ Y) {
  extern __shared__ float lds[];
  float* smA = lds;          // 32 x 516
  float* smB = lds + 16512;  // 128 x 68
  float* smG = lds + 25216;  // 32 x 132  (G tile)
  float* smX = lds + 29440;  // 128 x 72  (x chunk, k-major)
  float* sqi = lds + 38656;  // 32
  float* si  = lds + 38688;  // 32
  float* rmi = lds + 38720;  // 32
  float* sqj = lds + 38752;  // 128
  float* sj  = lds + 38880;  // 128  (total 39008 floats = 156,032 B)

  const int t = threadIdx.x;
  const int wave = t >> 5, lane = t & 31;
  const int half = lane >> 4, ln = lane & 15;
  const int i0 = blockIdx.x * 32;

  for (int idx = t; idx < 32 * 128; idx += 256) {
    const int r = idx >> 7, c = idx & 127;
    async_copy_b128(smA + r * 516 + c * 4, S + (size_t)(i0 + r) * KDIM + c * 4);
  }
  if (t < 32) {
    sqi[t] = sq[i0 + t];
    si[t] = ssum[i0 + t];
    rmi[t] = 1.0f / rowmax[i0 + t];
  }
  wait_async();

  const int strip = wave & 1;
  const int m_base = strip * 16;
  const int c0 = wave >> 1;  // gram col tiles c0, c0+4 ; G@x col tile c0

  v8f acc_y[8] = {};  // one 16x16 accumulator per 64-col chunk of x

  for (int j0 = 0; j0 < NROW; j0 += 128) {
    __syncthreads();
    if (t < 128) { sqj[t] = sq[j0 + t]; sj[t] = ssum[j0 + t]; }

    // ---- gram: rows [i0,32) x cols [j0,128) ----
    v8f g0 = {}, g1 = {};
    for (int kc = 0; kc < KDIM; kc += 64) {
      __syncthreads();
      for (int idx = t; idx < 128 * 16; idx += 256) {
        const int r = idx >> 4, c = idx & 15;
        async_copy_b128(smB + r * 68 + c * 4,
                        S + (size_t)(j0 + r) * KDIM + kc + c * 4);
      }
      wait_async();
      __syncthreads();
#pragma unroll
      for (int k = 0; k < 64; k += 4) {
        const v2f a  = frag_rm(smA, 516, m_base, kc + k);
        const v2f b0 = frag_rm(smB, 68, c0 * 16, k);
        const v2f b1 = frag_rm(smB, 68, (c0 + 4) * 16, k);
        g0 = wmma_f32(a, b0, g0);
        g1 = wmma_f32(a, b1, g1);
      }
    }

    // ---- elementwise: G = 1 - D/rowmax, into LDS ----
    const int n0g = c0 * 16 + ln;
    const int n1g = (c0 + 4) * 16 + ln;
    const float bsq0 = sqj[n0g], bs0 = sj[n0g];
    const float bsq1 = sqj[n1g], bs1 = sj[n1g];
#pragma unroll
    for (int v = 0; v < 8; ++v) {
      const int m = m_base + v + 8 * half;
      const float asq = sqi[m], as = si[m], inv = rmi[m];
      smG[m * 132 + n0g] = 1.0f - dist_val(asq, as, bsq0, bs0, g0[v]) * inv;
      smG[m * 132 + n1g] = 1.0f - dist_val(asq, as, bsq1, bs1, g1[v]) * inv;
    }

    // ---- accumulate G(32x128) @ x(128x512) in 64-col chunks ----
    for (int cc = 0; cc < 8; ++cc) {
      __syncthreads();  // smG ready for all / smX reusable
      for (int idx = t; idx < 128 * 16; idx += 256) {
        const int r = idx >> 4, c = idx & 15;
        async_copy_b128(smX + r * 72 + c * 4,
                        X + (size_t)(j0 + r) * KDIM + cc * 64 + c * 4);
      }
      wait_async();
      __syncthreads();
      v8f acc = acc_y[cc];
#pragma unroll
      for (int k = 0; k < 128; k += 4) {
        const v2f a = frag_rm(smG, 132, m_base, k);
        const v2f b = frag_km(smX, 72, k, c0 * 16);
        acc = wmma_f32(a, b, acc);
      }
      acc_y[cc] = acc;
    }
  }

#pragma unroll
  for (int cc = 0; cc < 8; ++cc) {
#pragma unroll
    for (int v = 0; v < 8; ++v) {
      const int row = i0 + m_base + v + 8 * half;
      const int col = cc * 64 + c0 * 16 + ln;
      Y[(size_t)row * KDIM + col] = acc_y[cc][v];
    }
  }
}

// ---------------------------------------------------------------------------
// Kernel 4: out = Y @ W  ([8192,512] x [512,512]) fp32 WMMA tiled GEMM.
// Block = 64 rows x 128 cols, K chunked by 64 through LDS.
// ---------------------------------------------------------------------------
__global__ __launch_bounds__(256) void out_gemm_kernel(
    const float* __restrict__ Y, const float* __restrict__ W,
    float* __restrict__ out) {
  __shared__ float smY[64 * 68];   // 17.4 KB
  __shared__ float smW[64 * 136];  // 34.8 KB
  const int t = threadIdx.x;
  const int wave = t >> 5, lane = t & 31;
  const int half = lane >> 4, ln = lane & 15;
  const int i0 = blockIdx.x * 64;
  const int n0 = blockIdx.y * 128;
  const int strip = wave & 3;  // 4 strips of 16 rows
  const int cb = wave >> 2;    // col tiles cb, cb+2, cb+4, cb+6

  v8f acc[4] = {};

  for (int kc = 0; kc < KDIM; kc += 64) {
    __syncthreads();
    for (int idx = t; idx < 64 * 16; idx += 256) {
      const int r = idx >> 4, c = idx & 15;
      async_copy_b128(smY + r * 68 + c * 4,
                      Y + (size_t)(i0 + r) * KDIM + kc + c * 4);
    }
    for (int idx = t; idx < 64 * 32; idx += 256) {
      const int r = idx >> 5, c = idx & 31;
      async_copy_b128(smW + r * 136 + c * 4,
                      W + (size_t)(kc + r) * KDIM + n0 + c * 4);
    }
    wait_async();
    __syncthreads();
#pragma unroll
    for (int k = 0; k < 64; k += 4) {
      const v2f a = frag_rm(smY, 68, strip * 16, k);
#pragma unroll
      for (int cc = 0; cc < 4; ++cc) {
        const v2f b = frag_km(smW, 136, k, (cb + 2 * cc) * 16);
        acc[cc] = wmma_f32(a, b, acc[cc]);
      }
    }
  }
#pragma unroll
  for (int cc = 0; cc < 4; ++cc) {
#pragma unroll
    for (int v = 0; v < 8; ++v) {
      const int row = i0 + strip * 16 + v + 8 * half;
      const int col = n0 + (cb + 2 * cc) * 16 + ln;
      out[(size_t)row * KDIM + col] = acc[cc][v];
    }
  }
}

// ---------------------------------------------------------------------------
extern "C" void kernel_launch(void* const* d_in, const int* in_sizes, int n_in,
                              void* d_out, int out_size, void* d_ws, size_t ws_size,
                              hipStream_t stream) {
  (void)in_sizes; (void)n_in; (void)out_size; (void)ws_size;
  const float* X = (const float*)d_in[0];      // x        [8192,512]
  const float* S = (const float*)d_in[1];      // sim_feat [8192,512]
  const float* W = (const float*)d_in[2];      // weight   [512,512]
  float* out = (float*)d_out;                  // [8192,512]

  float* ws = (float*)d_ws;                    // ~16.9 MB used
  float* sq   = ws;                            // [8192]
  float* ssum = ws + NROW;                     // [8192]
  float* rmax = ws + 2 * NROW;                 // [8192]
  float* Yb   = ws + 3 * NROW;                 // [8192,512]

  stats_init_kernel<<<NROW / 8, 256, 0, stream>>>(S, sq, ssum, rmax);
  rowmax_kernel<<<NROW / 32, 256, 25536 * sizeof(float), stream>>>(S, sq, ssum, rmax);
  fused_gx_kernel<<<NROW / 32, 256, 39008 * sizeof(float), stream>>>(S, X, sq, ssum, rmax, Yb);
  out_gemm_kernel<<<dim3(NROW / 64, 4), 256, 0, stream>>>(Yb, W, out);
}